// BGguidedLoss_49813030699408
// MI455X (gfx1250) — compile-verified
//
#include <hip/hip_runtime.h>

typedef __attribute__((ext_vector_type(2))) float v2f;
typedef __attribute__((ext_vector_type(8))) float v8f;

#define NBLOCKS_MAX 4096
#define NTHREADS 256

// --- rgb -> (H, V) only (skimage-style), matching jnp.where priority r,g,b ---
__device__ __forceinline__ void rgb_to_hv(float r, float g, float b,
                                          float& h, float& v) {
  float maxc = fmaxf(r, fmaxf(g, b));
  float minc = fminf(r, fminf(g, b));
  float delta = maxc - minc;
  float sd = (delta == 0.0f) ? 1.0f : delta;
  float inv = 1.0f / sd;
  float hh = (maxc == r) ? (g - b) * inv
           : (maxc == g) ? 2.0f + (b - r) * inv
                         : 4.0f + (r - g) * inv;
  hh *= 0.16666666666666666f;   // h / 6
  hh -= floorf(hh);             // python-style mod 1.0
  h = (delta == 0.0f) ? 0.0f : hh;
  v = maxc;                     // V channel == maxc
}

__device__ __forceinline__ float loss_full(float gr, float gg, float gb,
                                           float br, float bgc, float bbc,
                                           float fr, float fgc, float fbc,
                                           float u, float threshold) {
  float hg, vg, hb, vb;
  rgb_to_hv(gr, gg, gb, hg, vg);
  rgb_to_hv(br, bgc, bbc, hb, vb);
  float dh = hg - hb, dv = vg - vb;
  float diff = sqrtf(dh * dh + dv * dv);
  // sigmoid((diff - threshold)/0.1)
  float mask = 1.0f / (1.0f + expf((threshold - diff) * 10.0f));
  float d0 = gr - br, d1 = gg - bgc, d2 = gb - bbc;
  float bg_mse = (d0 * d0 + d1 * d1 + d2 * d2) * (1.0f / 3.0f);
  float e0 = gr - fr, e1 = gg - fgc, e2 = gb - fbc;
  float fg_mse = (e0 * e0 + e1 * e1 + e2 * e2) * (1.0f / 3.0f);
  float per = fg_mse / (2.0f * u * u) + logf(u);
  return bg_mse * (1.0f - mask) + per * mask;
}

__device__ __forceinline__ float loss_simple(float gr, float gg, float gb,
                                             float br, float bgc, float bbc) {
  float d0 = gr - br, d1 = gg - bgc, d2 = gb - bbc;
  return d0 * d0 + d1 * d1 + d2 * d2;   // scaled by 1/(3N) at the end
}

// Stage 1: streaming pass, 4 rays/thread/iter via b128 loads; block partials.
__global__ __launch_bounds__(NTHREADS) void loss_main_kernel(
    const float* __restrict__ gt, const float* __restrict__ bg,
    const float* __restrict__ fg, const float* __restrict__ unc,
    const float* __restrict__ tp_ptr, const int* __restrict__ iter_ptr,
    float* __restrict__ partials, int N) {
  const float tp = tp_ptr[0];
  const int iter = iter_ptr[0];
  const float threshold = 1.414f * (1.0f - 1.0f / (1.0f + expf(-tp)));

  const int tid = blockIdx.x * blockDim.x + threadIdx.x;
  const int nth = gridDim.x * blockDim.x;
  const int ngroups = N >> 2;

  float acc = 0.0f;

  for (int g4 = tid; g4 < ngroups; g4 += nth) {
    const size_t base12 = 12u * (size_t)g4;
    const float4* gp = (const float4*)(gt + base12);
    const float4* bp = (const float4*)(bg + base12);
    float4 g0 = gp[0], g1 = gp[1], g2 = gp[2];
    float4 b0 = bp[0], b1 = bp[1], b2 = bp[2];
    // Unpack 4 rays x 3 channels from 3 float4s
    float GR[4] = {g0.x, g0.w, g1.z, g2.y};
    float GG[4] = {g0.y, g1.x, g1.w, g2.z};
    float GB[4] = {g0.z, g1.y, g2.x, g2.w};
    float BR[4] = {b0.x, b0.w, b1.z, b2.y};
    float BGc[4] = {b0.y, b1.x, b1.w, b2.z};
    float BB[4] = {b0.z, b1.y, b2.x, b2.w};
    if (iter > 300) {
      const float4* fp = (const float4*)(fg + base12);
      float4 f0 = fp[0], f1 = fp[1], f2 = fp[2];
      float FR[4] = {f0.x, f0.w, f1.z, f2.y};
      float FGc[4] = {f0.y, f1.x, f1.w, f2.z};
      float FB[4] = {f0.z, f1.y, f2.x, f2.w};
      float4 u4 = *(const float4*)(unc + 4u * (size_t)g4);
      float U[4] = {u4.x, u4.y, u4.z, u4.w};
#pragma unroll
      for (int k = 0; k < 4; ++k)
        acc += loss_full(GR[k], GG[k], GB[k], BR[k], BGc[k], BB[k],
                         FR[k], FGc[k], FB[k], U[k], threshold);
    } else {
#pragma unroll
      for (int k = 0; k < 4; ++k)
        acc += loss_simple(GR[k], GG[k], GB[k], BR[k], BGc[k], BB[k]);
    }
  }

  // Tail rays (N % 4) handled by global thread 0 only.
  if (tid == 0) {
    for (int i = ngroups << 2; i < N; ++i) {
      const size_t o = 3u * (size_t)i;
      if (iter > 300)
        acc += loss_full(gt[o], gt[o + 1], gt[o + 2],
                         bg[o], bg[o + 1], bg[o + 2],
                         fg[o], fg[o + 1], fg[o + 2], unc[i], threshold);
      else
        acc += loss_simple(gt[o], gt[o + 1], gt[o + 2],
                           bg[o], bg[o + 1], bg[o + 2]);
    }
  }

  __shared__ float sm[NTHREADS];
  sm[threadIdx.x] = acc;
  __syncthreads();
#pragma unroll
  for (int s = NTHREADS / 2; s > 0; s >>= 1) {
    if (threadIdx.x < s) sm[threadIdx.x] += sm[threadIdx.x + s];
    __syncthreads();
  }
  if (threadIdx.x == 0) partials[blockIdx.x] = sm[0];
}

// Stage 2: one wave32. Strided partial sums, then V_WMMA_F32_16X16X4_F32
// cross-lane reduction: A = lane partials (16x4 layout: lanes 0-15 -> K0,
// lanes 16-31 -> K2), B = all-ones => D rows are row-sums of A. Summing the
// 8 D VGPRs per lane gives half-totals; one shfl_xor(16) completes the sum.
__global__ __launch_bounds__(32) void loss_reduce_kernel(
    const float* __restrict__ partials, int n_partials,
    const int* __restrict__ iter_ptr, float* __restrict__ out,
    float invN, float inv3N) {
  const int lane = threadIdx.x;
  float s = 0.0f;
  for (int i = lane; i < n_partials; i += 32) s += partials[i];

  v2f a;    a[0] = s;    a[1] = 0.0f;
  v2f ones; ones[0] = 1.0f; ones[1] = 1.0f;
  v8f c = {};
  v8f d = __builtin_amdgcn_wmma_f32_16x16x4_f32(
      /*neg_a=*/false, a, /*neg_b=*/false, ones,
      /*c_mod=*/(short)0, c, /*reuse_a=*/false, /*reuse_b=*/false);
  float t = d[0] + d[1] + d[2] + d[3] + d[4] + d[5] + d[6] + d[7];
  t += __shfl_xor(t, 16, 32);   // lanes <16 hold rows 0-7, lanes >=16 rows 8-15

  if (lane == 0) {
    const int iter = iter_ptr[0];
    out[0] = t * ((iter > 300) ? invN : inv3N);
  }
}

extern "C" void kernel_launch(void* const* d_in, const int* in_sizes, int n_in,
                              void* d_out, int out_size, void* d_ws, size_t ws_size,
                              hipStream_t stream) {
  const float* gt  = (const float*)d_in[0];
  const float* bg  = (const float*)d_in[1];
  const float* fg  = (const float*)d_in[2];
  // d_in[3] = FG_acc (unused by reference loss path)
  const float* unc = (const float*)d_in[4];
  const float* tp  = (const float*)d_in[5];
  // d_in[6] = steepness (unused; reference hardcodes 0.1)
  const int* iter  = (const int*)d_in[7];

  const int N = in_sizes[0] / 3;

  int blocks = NBLOCKS_MAX;
  int max_ws = (int)(ws_size / sizeof(float));
  if (max_ws < blocks) blocks = (max_ws > 0) ? max_ws : 1;
  float* partials = (float*)d_ws;

  loss_main_kernel<<<blocks, NTHREADS, 0, stream>>>(gt, bg, fg, unc, tp, iter,
                                                    partials, N);
  const float invN = 1.0f / (float)N;
  loss_reduce_kernel<<<1, 32, 0, stream>>>(partials, blocks, iter,
                                           (float*)d_out, invN, invN / 3.0f);
}